// LocalSelfAttention_10849087390046
// MI455X (gfx1250) — compile-verified
//
#include <hip/hip_runtime.h>
#include <math.h>

// ---------------------------------------------------------------------------
// Local self-attention for MI455X (gfx1250), all matmuls via V_WMMA_F32_16X16X4_F32
//   B=4, T=4096, C=256, H=4, D=64, window=25 (half-window 12)
// Pipeline:
//   k1: qkv = x @ W_qkv + b   -> scatter to q(*1/8)/k/v in [B,H,T,D]
//       (each wave: 64x16 output strip = 4 M-subtiles sharing one B fragment)
//   k2: per (b,h,qtile16): S = qK^T over 3 key tiles, masked softmax in
//       C-fragment layout, P -> LDS -> A-fragment, O = P V, write [B,T,C]
//   k3: out = attn @ W_out + b_out (same 64x16-per-wave strip shape)
// ---------------------------------------------------------------------------

typedef float v2f __attribute__((ext_vector_type(2)));
typedef float v8f __attribute__((ext_vector_type(8)));

#define T_SEQ   4096
#define C_HID   256
#define N_HEAD  4
#define D_HEAD  64
#define HALF_W  12

static __device__ __forceinline__ v8f wmma_f32(v2f a, v2f b, v8f c) {
    // D = A(16x4) * B(4x16) + C(16x16), fp32, wave32
    return __builtin_amdgcn_wmma_f32_16x16x4_f32(
        /*neg_a=*/false, a, /*neg_b=*/false, b,
        /*c_mod=*/(short)0, c, /*reuse_a=*/false, /*reuse_b=*/false);
}

// ---------------------------------------------------------------------------
// Kernel 1: QKV projection.  x:[B*T,256] @ W:[256,768] + bias.
// One wave per 64x16 output strip (4 stacked 16x16 WMMA tiles reusing the
// same weight fragment). Epilogue scatters into q/k/v [B,H,T,D]; q * 0.125.
// ---------------------------------------------------------------------------
__global__ __launch_bounds__(128) void qkv_kernel(
    const float* __restrict__ x, const float* __restrict__ Wqkv,
    const float* __restrict__ bqkv,
    float* __restrict__ qb, float* __restrict__ kb, float* __restrict__ vb)
{
    const int lane  = threadIdx.x & 31;
    const int wave  = threadIdx.x >> 5;
    const int strip = blockIdx.x * 4 + wave;         // 256*48 strips
    const int NT    = (3 * C_HID) / 16;              // 48
    const int mt4   = strip / NT;                    // 64-row supertile
    const int nt    = strip % NT;

    const int half = lane >> 4;                      // 0 / 1
    const int lm   = lane & 15;

    const int bcol = nt * 16 + lm;                   // W column

    const float* ap0 = x + (size_t)(mt4 * 64 +  0 + lm) * C_HID + 2 * half;
    const float* ap1 = x + (size_t)(mt4 * 64 + 16 + lm) * C_HID + 2 * half;
    const float* ap2 = x + (size_t)(mt4 * 64 + 32 + lm) * C_HID + 2 * half;
    const float* ap3 = x + (size_t)(mt4 * 64 + 48 + lm) * C_HID + 2 * half;
    const float* bp  = Wqkv + (size_t)(2 * half) * (3 * C_HID) + bcol;

    v8f acc0 = {}, acc1 = {}, acc2 = {}, acc3 = {};
    #pragma unroll 2
    for (int k0 = 0; k0 < C_HID; k0 += 4) {
        v2f b; b.x = bp[0]; b.y = bp[3 * C_HID];     // W[k0+2h][col], W[k0+2h+1][col]
        v2f a0 = *(const v2f*)ap0;
        v2f a1 = *(const v2f*)ap1;
        v2f a2 = *(const v2f*)ap2;
        v2f a3 = *(const v2f*)ap3;
        acc0 = wmma_f32(a0, b, acc0);
        acc1 = wmma_f32(a1, b, acc1);
        acc2 = wmma_f32(a2, b, acc2);
        acc3 = wmma_f32(a3, b, acc3);
        ap0 += 4; ap1 += 4; ap2 += 4; ap3 += 4;
        bp += 4 * (3 * C_HID);
    }

    const float bias  = bqkv[bcol];
    const int   which = bcol >> 8;                   // 0=q 1=k 2=v
    const int   h     = (bcol >> 6) & 3;
    const int   d     = bcol & 63;
    float* outp = (which == 0) ? qb : (which == 1) ? kb : vb;
    const float scale = (which == 0) ? 0.125f : 1.0f;

    v8f accs[4] = {acc0, acc1, acc2, acc3};
    #pragma unroll
    for (int s = 0; s < 4; ++s) {
        #pragma unroll
        for (int r = 0; r < 8; ++r) {
            const int row = mt4 * 64 + s * 16 + r + 8 * half;  // b*T + t
            const int b_  = row >> 12;
            const int t   = row & (T_SEQ - 1);
            const float v = (accs[s][r] + bias) * scale;
            outp[(((size_t)(b_ * N_HEAD + h)) * T_SEQ + t) * D_HEAD + d] = v;
        }
    }
}

// ---------------------------------------------------------------------------
// Kernel 2: sliding-window attention. One wave per (b, h, 16-query tile).
// Window 25 -> keys span exactly 3 aligned 16-key tiles [t0-16, t0+32).
// ---------------------------------------------------------------------------
__global__ __launch_bounds__(128) void attn_kernel(
    const float* __restrict__ qb, const float* __restrict__ kb,
    const float* __restrict__ vb, float* __restrict__ attn /* [B,T,C] */)
{
    __shared__ float plds[4][16 * 50];               // per-wave P tile, stride 50
    const int lane = threadIdx.x & 31;
    const int wave = threadIdx.x >> 5;
    const int wid  = blockIdx.x * 4 + wave;          // 0..4095
    const int qt   = wid & (T_SEQ / 16 - 1);         // 256 q-tiles
    const int h    = (wid >> 8) & (N_HEAD - 1);
    const int b    = wid >> 10;

    const int half = lane >> 4;
    const int lm   = lane & 15;
    const int t0   = qt * 16;

    const size_t base = ((size_t)(b * N_HEAD + h)) * T_SEQ * D_HEAD;
    const float* Q = qb + base;                      // q already scaled by 1/8
    const float* K = kb + base;
    const float* V = vb + base;

    // ---- scores: S[kt] = Q Ktile^T, kt in {-1,0,+1} relative tiles ----
    v8f s[3];
    #pragma unroll
    for (int kt = 0; kt < 3; ++kt) {
        int keyrow = t0 + (kt - 1) * 16 + lm;
        int keyc   = keyrow < 0 ? 0 : (keyrow > T_SEQ - 1 ? T_SEQ - 1 : keyrow);
        const float* ap = Q + (size_t)(t0 + lm) * D_HEAD + 2 * half;  // A: Q[q][kd]
        const float* bp = K + (size_t)keyc      * D_HEAD + 2 * half;  // B: K^T[kd][key]
        v8f acc = {};
        #pragma unroll
        for (int k0 = 0; k0 < D_HEAD; k0 += 4) {
            v2f a  = *(const v2f*)ap;
            v2f bb = *(const v2f*)bp;
            acc = wmma_f32(a, bb, acc);
            ap += 4; bp += 4;
        }
        s[kt] = acc;
    }

    // ---- masked softmax in C-fragment layout (row m=r+8*half, col lm) ----
    float p[3][8];
    float rsum[8];
    #pragma unroll
    for (int r = 0; r < 8; ++r) {
        const int q = t0 + r + 8 * half;
        float mx = -1e30f;
        #pragma unroll
        for (int kt = 0; kt < 3; ++kt) {
            const int kcol = t0 + (kt - 1) * 16 + lm;
            const int dd   = q - kcol;
            const bool ok  = (kcol >= 0) && (kcol < T_SEQ) &&
                             (dd >= -HALF_W) && (dd <= HALF_W);
            p[kt][r] = ok ? s[kt][r] : -1e30f;
            mx = fmaxf(mx, p[kt][r]);
        }
        // row max / sum: reduce across the 16 lanes of this half-group
        #pragma unroll
        for (int off = 1; off < 16; off <<= 1)
            mx = fmaxf(mx, __shfl_xor(mx, off, 32));
        float sum = 0.0f;
        #pragma unroll
        for (int kt = 0; kt < 3; ++kt) {
            const float e = (p[kt][r] > -1e29f) ? __expf(p[kt][r] - mx) : 0.0f;
            p[kt][r] = e;
            sum += e;
        }
        #pragma unroll
        for (int off = 1; off < 16; off <<= 1)
            sum += __shfl_xor(sum, off, 32);
        rsum[r] = sum;                               // > 0: window includes self
    }

    // ---- C-layout -> A-layout via per-wave LDS tile (no barrier needed:
    //      LDS ops from one wave are in-order) ----
    float* pl = plds[wave];
    #pragma unroll
    for (int r = 0; r < 8; ++r) {
        const int m = r + 8 * half;
        #pragma unroll
        for (int kt = 0; kt < 3; ++kt)
            pl[m * 50 + kt * 16 + lm] = p[kt][r];
    }
    asm volatile("" ::: "memory");                   // keep stores before loads

    // ---- O = P(16x48) * V(48x64) ----
    #pragma unroll
    for (int dt = 0; dt < 4; ++dt) {
        v8f acc = {};
        #pragma unroll
        for (int k0 = 0; k0 < 48; k0 += 4) {
            const int kk = k0 + 2 * half;
            // A: P[m=lm][kk..kk+1] from LDS (8B aligned: 50*lm + kk even)
            v2f a = *(const v2f*)&pl[lm * 50 + kk];
            int row0 = t0 - 16 + kk;
            int r0 = row0 < 0 ? 0 : (row0 > T_SEQ - 1 ? T_SEQ - 1 : row0);
            int r1 = row0 + 1 < 0 ? 0 : (row0 + 1 > T_SEQ - 1 ? T_SEQ - 1 : row0 + 1);
            v2f bb;                                   // B: V[kk][d], V[kk+1][d]
            bb.x = V[(size_t)r0 * D_HEAD + dt * 16 + lm];
            bb.y = V[(size_t)r1 * D_HEAD + dt * 16 + lm];
            acc = wmma_f32(a, bb, acc);
        }
        // normalize + write to [B,T,C] (row mapping identical to rsum's)
        #pragma unroll
        for (int r = 0; r < 8; ++r) {
            const int t = t0 + r + 8 * half;
            const float v = acc[r] / rsum[r];
            attn[((size_t)(b * T_SEQ + t)) * C_HID + h * D_HEAD + dt * 16 + lm] = v;
        }
    }
}

// ---------------------------------------------------------------------------
// Kernel 3: output projection. attn:[B*T,256] @ W_out:[256,256] + b_out.
// One wave per 64x16 strip (4 M-subtiles sharing the B fragment).
// ---------------------------------------------------------------------------
__global__ __launch_bounds__(128) void proj_kernel(
    const float* __restrict__ a_in, const float* __restrict__ Wout,
    const float* __restrict__ bout, float* __restrict__ out)
{
    const int lane  = threadIdx.x & 31;
    const int wave  = threadIdx.x >> 5;
    const int strip = blockIdx.x * 4 + wave;         // 256*16 strips
    const int NT    = C_HID / 16;                    // 16
    const int mt4   = strip / NT;
    const int nt    = strip % NT;

    const int half = lane >> 4;
    const int lm   = lane & 15;

    const int bcol = nt * 16 + lm;

    const float* ap0 = a_in + (size_t)(mt4 * 64 +  0 + lm) * C_HID + 2 * half;
    const float* ap1 = a_in + (size_t)(mt4 * 64 + 16 + lm) * C_HID + 2 * half;
    const float* ap2 = a_in + (size_t)(mt4 * 64 + 32 + lm) * C_HID + 2 * half;
    const float* ap3 = a_in + (size_t)(mt4 * 64 + 48 + lm) * C_HID + 2 * half;
    const float* bp  = Wout + (size_t)(2 * half) * C_HID + bcol;

    v8f acc0 = {}, acc1 = {}, acc2 = {}, acc3 = {};
    #pragma unroll 2
    for (int k0 = 0; k0 < C_HID; k0 += 4) {
        v2f b; b.x = bp[0]; b.y = bp[C_HID];
        v2f a0 = *(const v2f*)ap0;
        v2f a1 = *(const v2f*)ap1;
        v2f a2 = *(const v2f*)ap2;
        v2f a3 = *(const v2f*)ap3;
        acc0 = wmma_f32(a0, b, acc0);
        acc1 = wmma_f32(a1, b, acc1);
        acc2 = wmma_f32(a2, b, acc2);
        acc3 = wmma_f32(a3, b, acc3);
        ap0 += 4; ap1 += 4; ap2 += 4; ap3 += 4;
        bp += 4 * C_HID;
    }

    const float bias = bout[bcol];
    v8f accs[4] = {acc0, acc1, acc2, acc3};
    #pragma unroll
    for (int s = 0; s < 4; ++s) {
        #pragma unroll
        for (int r = 0; r < 8; ++r) {
            const int row = mt4 * 64 + s * 16 + r + 8 * half;
            out[(size_t)row * C_HID + bcol] = accs[s][r] + bias;
        }
    }
}

// ---------------------------------------------------------------------------
extern "C" void kernel_launch(void* const* d_in, const int* in_sizes, int n_in,
                              void* d_out, int out_size, void* d_ws, size_t ws_size,
                              hipStream_t stream) {
    const float* x     = (const float*)d_in[0];   // [4,4096,256]
    const float* W_qkv = (const float*)d_in[1];   // [256,768]
    const float* b_qkv = (const float*)d_in[2];   // [768]
    const float* W_out = (const float*)d_in[3];   // [256,256]
    const float* b_out = (const float*)d_in[4];   // [256]
    float* out = (float*)d_out;                   // [4,4096,256]

    const size_t NPER = (size_t)4 * N_HEAD * T_SEQ * D_HEAD; // 4,194,304 floats
    float* qb  = (float*)d_ws;
    float* kb  = qb + NPER;
    float* vb  = kb + NPER;
    float* att = vb + NPER;                       // [B,T,C] = NPER floats

    // 256*48 strips / 4 waves per block
    qkv_kernel <<<3072, 128, 0, stream>>>(x, W_qkv, b_qkv, qb, kb, vb);
    // 4*4*256 waves / 4 waves per block
    attn_kernel<<<1024, 128, 0, stream>>>(qb, kb, vb, att);
    // 256*16 strips / 4 waves per block
    proj_kernel<<<1024, 128, 0, stream>>>(att, W_out, b_out, out);
}